// GNN_47725676593438
// MI455X (gfx1250) — compile-verified
//
#include <hip/hip_runtime.h>
#include <hip/hip_bf16.h>

#define NNODES  50000
#define NEDGES  800000
#define IN_DIM  256
#define OUT_DIM 128
#define NCLS    40

typedef __attribute__((ext_vector_type(2))) float v2f;
typedef __attribute__((ext_vector_type(8))) float v8f;
typedef __attribute__((ext_vector_type(4))) int   v4i;

#if defined(__AMDGCN__)
typedef __attribute__((address_space(1))) v4i* g_v4i;   // global (AS1)
typedef __attribute__((address_space(3))) v4i* l_v4i;   // LDS (AS3)
#endif

static __device__ __forceinline__ void atomic_add_f32(float* p, float v) {
    __hip_atomic_fetch_add(p, v, __ATOMIC_RELAXED, __HIP_MEMORY_SCOPE_AGENT);
}

static __device__ __forceinline__ void wait_async_all() {
#if __has_builtin(__builtin_amdgcn_s_wait_asynccnt)
    __builtin_amdgcn_s_wait_asynccnt(0);
#else
    asm volatile("s_wait_asynccnt 0x0" ::: "memory");
#endif
}

// ---------------------------------------------------------------- zero fill
__global__ __launch_bounds__(256) void k_zero(float* __restrict__ p, int n) {
    int i = blockIdx.x * blockDim.x + threadIdx.x;
    int stride = gridDim.x * blockDim.x;
    for (; i < n; i += stride) p[i] = 0.0f;
}

// ---------------------------------------------------------------- degrees
__global__ __launch_bounds__(256) void k_degree(const int* __restrict__ src,
                                                const int* __restrict__ dst,
                                                float* __restrict__ deg_src,
                                                float* __restrict__ deg_dst) {
    int e = blockIdx.x * blockDim.x + threadIdx.x;
    if (e < NEDGES) {
        atomic_add_f32(&deg_src[src[e]], 1.0f);
        atomic_add_f32(&deg_dst[dst[e]], 1.0f);
    }
}

// deg -> rsqrt(max(deg,1)) in place (both deg arrays are contiguous)
__global__ __launch_bounds__(256) void k_norm(float* __restrict__ deg, int n) {
    int i = blockIdx.x * blockDim.x + threadIdx.x;
    if (i < n) {
        float d = deg[i];
        deg[i] = rsqrtf(fmaxf(d, 1.0f));
    }
}

// ---------------------------------------------------------------- GEMM1:
// h[m, n] = sum_k (emb[nodes[m], k] * norm_src[m]) * W[k, n]
// block = 256 threads (8 waves), one 16-row tile per block, wave w -> cols [16w,16w+16)
// A-tile staged to LDS via CDNA5 async-to-LDS (ASYNCcnt); norm_src folded at A-read.
__global__ __launch_bounds__(256) void k_gemm1(const int*   __restrict__ nodes,
                                               const float* __restrict__ emb,
                                               const float* __restrict__ W,
                                               const float* __restrict__ norm_src,
                                               float*       __restrict__ h) {
    // pad row stride to 260 floats: 260 % 64 == 4 -> conflict-free b64 A reads
    __shared__ float As[16][IN_DIM + 4];

    const int m0 = blockIdx.x * 16;
    const int t  = threadIdx.x;

    // stage 16x256 A tile (un-scaled): 16 contiguous floats / thread
    {
        const int r  = t >> 4;          // 0..15
        const int c0 = (t & 15) * 16;   // 0..240
        const int node = nodes[m0 + r];
        float* erow = (float*)(emb + (size_t)node * IN_DIM + c0);
        float* lrow = &As[r][c0];
#if __has_builtin(__builtin_amdgcn_global_load_async_to_lds_b128)
        // 4 x 16B async copies, global -> LDS, no VGPR round-trip (ASYNCcnt path)
        __builtin_amdgcn_global_load_async_to_lds_b128((g_v4i)(erow + 0),  (l_v4i)(lrow + 0),  0, 0);
        __builtin_amdgcn_global_load_async_to_lds_b128((g_v4i)(erow + 4),  (l_v4i)(lrow + 4),  0, 0);
        __builtin_amdgcn_global_load_async_to_lds_b128((g_v4i)(erow + 8),  (l_v4i)(lrow + 8),  0, 0);
        __builtin_amdgcn_global_load_async_to_lds_b128((g_v4i)(erow + 12), (l_v4i)(lrow + 12), 0, 0);
        wait_async_all();
#else
        #pragma unroll
        for (int j = 0; j < 16; ++j) lrow[j] = erow[j];
#endif
    }
    __syncthreads();

    const int lane = t & 31;
    const int wave = t >> 5;            // 0..7
    const int n0   = wave * 16;
    const int half = lane >> 4;         // 0: K=k,k+1   1: K=k+2,k+3
    const int lr   = lane & 15;
    const float s  = norm_src[m0 + lr]; // this lane always feeds A-row `lr`

    v8f c = {};
    #pragma unroll 4
    for (int k0 = 0; k0 < IN_DIM; k0 += 4) {
        const int k = k0 + 2 * half;
        // A: 16x4 f32 layout (ISA 7.12.2): lane<16 holds K=k0,k0+1; lane>=16 K=k0+2,k0+3
        v2f a = *(const v2f*)&As[lr][k];            // 8B aligned (k even, stride even)
        a[0] *= s;
        a[1] *= s;
        v2f bb;
        bb[0] = W[(k + 0) * OUT_DIM + n0 + lr];     // B: 4x16, row striped across lanes
        bb[1] = W[(k + 1) * OUT_DIM + n0 + lr];
        c = __builtin_amdgcn_wmma_f32_16x16x4_f32(false, a, false, bb,
                                                  (short)0, c, false, false);
    }

    // C/D layout: VGPR r -> M = r + 8*half, N = lr
    #pragma unroll
    for (int r = 0; r < 8; ++r) {
        const int row = m0 + r + half * 8;
        h[(size_t)row * OUT_DIM + n0 + lr] = c[r];
    }
}

// ---------------------------------------------------------------- SpMM:
// agg[dst[e], :] += h[src[e], :]   (one wave = one edge; lane l -> floats [4l,4l+4))
// edge endpoints forced wave-uniform -> scalar s_load instead of 32x redundant VMEM
__global__ __launch_bounds__(256) void k_spmm(const int*   __restrict__ src,
                                              const int*   __restrict__ dst,
                                              const float* __restrict__ h,
                                              float*       __restrict__ agg) {
    int e = blockIdx.x * (256 / 32) + (threadIdx.x >> 5);
    e = __builtin_amdgcn_readfirstlane(e);          // provably wave-uniform
    const int c4 = (threadIdx.x & 31) << 2;         // 0,4,...,124
    if (e < NEDGES) {
        const int s = src[e];                       // scalar loads (uniform)
        const int d = dst[e];
        const float4 v = *(const float4*)(h + (size_t)s * OUT_DIM + c4);
        float* dp = agg + (size_t)d * OUT_DIM + c4;
        atomic_add_f32(dp + 0, v.x);
        atomic_add_f32(dp + 1, v.y);
        atomic_add_f32(dp + 2, v.z);
        atomic_add_f32(dp + 3, v.w);
    }
}

// ---------------------------------------------------------------- epilogue:
// feat_out = agg*norm_dst + b  (stored);  logits = feat_out @ mlp_w + mlp_b
// block = 128 threads (4 waves); waves 0..2 cover N-tiles 0..47 (cols >= 40 masked)
__global__ __launch_bounds__(128) void k_out(const float* __restrict__ agg,
                                             const float* __restrict__ norm_dst,
                                             const float* __restrict__ bvec,
                                             const float* __restrict__ mlp_w,
                                             const float* __restrict__ mlp_b,
                                             float*       __restrict__ feat_out,
                                             float*       __restrict__ logits) {
    __shared__ float As[16][OUT_DIM + 4];    // stride 132 -> conflict-free b64 reads

    const int m0 = blockIdx.x * 16;
    const int t  = threadIdx.x;

    // stage 16x128 tile of feat_out = agg*norm + b; also write it to d_out
    {
        const int r  = t >> 3;          // 0..15 (8 threads/row)
        const int c0 = (t & 7) * 16;
        const float s = norm_dst[m0 + r];
        #pragma unroll
        for (int j = 0; j < 16; ++j) {
            const float v = agg[(size_t)(m0 + r) * OUT_DIM + c0 + j] * s + bvec[c0 + j];
            As[r][c0 + j] = v;
            feat_out[(size_t)(m0 + r) * OUT_DIM + c0 + j] = v;
        }
    }
    __syncthreads();

    const int lane = t & 31;
    const int wave = t >> 5;
    if (wave < 3) {                      // wave-uniform branch; EXEC all-ones inside
        const int n0   = wave * 16;
        const int half = lane >> 4;
        const int lr   = lane & 15;
        const int n    = n0 + lr;
        const bool nok = (n < NCLS);

        v8f c = {};
        #pragma unroll 4
        for (int k0 = 0; k0 < OUT_DIM; k0 += 4) {
            const int k = k0 + 2 * half;
            v2f a = *(const v2f*)&As[lr][k];
            v2f bb;
            bb[0] = nok ? mlp_w[(k + 0) * NCLS + n] : 0.0f;   // select, no EXEC change
            bb[1] = nok ? mlp_w[(k + 1) * NCLS + n] : 0.0f;
            c = __builtin_amdgcn_wmma_f32_16x16x4_f32(false, a, false, bb,
                                                      (short)0, c, false, false);
        }

        if (nok) {                        // divergence only after the last WMMA
            const float mb = mlp_b[n];
            #pragma unroll
            for (int r = 0; r < 8; ++r) {
                const int row = m0 + r + half * 8;
                logits[(size_t)row * NCLS + n] = c[r] + mb;
            }
        }
    }
}

// ---------------------------------------------------------------- launch
extern "C" void kernel_launch(void* const* d_in, const int* in_sizes, int n_in,
                              void* d_out, int out_size, void* d_ws, size_t ws_size,
                              hipStream_t stream) {
    const int*   nodes = (const int*)  d_in[0];
    const int*   src   = (const int*)  d_in[1];
    const int*   dst   = (const int*)  d_in[2];
    const float* emb   = (const float*)d_in[3];
    const float* W     = (const float*)d_in[4];
    const float* bvec  = (const float*)d_in[5];
    const float* mlp_w = (const float*)d_in[6];
    const float* mlp_b = (const float*)d_in[7];

    float* out      = (float*)d_out;
    float* feat_out = out;                                  // 50000*128
    float* logits   = out + (size_t)NNODES * OUT_DIM;       // 50000*40

    // workspace layout: [deg_src | deg_dst | agg | h]  (zeroed region is contiguous)
    float* ws       = (float*)d_ws;
    float* deg_src  = ws;
    float* deg_dst  = ws + NNODES;
    float* agg      = ws + 2 * NNODES;
    float* h        = agg + (size_t)NNODES * OUT_DIM;

    const int zero_n = 2 * NNODES + NNODES * OUT_DIM;       // degs + agg
    k_zero<<<2048, 256, 0, stream>>>(ws, zero_n);

    k_degree<<<(NEDGES + 255) / 256, 256, 0, stream>>>(src, dst, deg_src, deg_dst);

    k_norm<<<(2 * NNODES + 255) / 256, 256, 0, stream>>>(ws, 2 * NNODES);

    k_gemm1<<<NNODES / 16, 256, 0, stream>>>(nodes, emb, W, deg_src, h);

    k_spmm<<<NEDGES / 8, 256, 0, stream>>>(src, dst, h, agg);

    k_out<<<NNODES / 16, 128, 0, stream>>>(agg, deg_dst, bvec, mlp_w, mlp_b,
                                           feat_out, logits);
}